// _FlaxMPS_32375463477809
// MI455X (gfx1250) — compile-verified
//
#include <hip/hip_runtime.h>
#include <cmath>

// MPS chain contraction for MI455X (gfx1250, wave32, WMMA).
// One wave per batch row; the 1024-site matrix chain is split into two
// independent 512-site half-chains (ILP: 8 WMMAs in flight per wave).
// carry kept TRANSPOSED as two WMMA B-operand fragments (v16h each);
// per site: 4x v_wmma_f16_16x16x32_f16 with loop-invariant A^T fragments,
// then half-wave lane swaps (VALU permlane, not LDS) re-form the B operands.
// Final: trace(C2*C1) via a one-time 2KB LDS round-trip + per-lane dot.

typedef _Float16 v16h __attribute__((ext_vector_type(16)));
typedef _Float16 v8h  __attribute__((ext_vector_type(8)));

#define MPS_BATCH  4096
#define MPS_NSITES 1024
#define MPS_BOND   32
#define MPS_HALFN  (MPS_NSITES / 2)
#define WAVES_PER_BLOCK 8

union U8 { v8h h; unsigned u[4]; };

__device__ __forceinline__ unsigned ballot32(bool p) {
#if __has_builtin(__builtin_amdgcn_ballot_w32)
    return __builtin_amdgcn_ballot_w32(p);
#else
    return (unsigned)__ballot(p);
#endif
}

// a' = { a[0:15] , b[0:15] -> lanes 16:31 }
// b' = { a[16:31] -> lanes 0:15 , b[16:31] }
__device__ __forceinline__ void swap_halves(unsigned &a, unsigned &b, int lane) {
#if __has_builtin(__builtin_amdgcn_permlane16_swap)
    typedef unsigned uv2 __attribute__((ext_vector_type(2)));
    uv2 r = __builtin_amdgcn_permlane16_swap(a, b, false, false);
    a = r[0]; b = r[1];
#elif __has_builtin(__builtin_amdgcn_permlanex16)
    // identity-position cross-half gather: lane i reads other half, pos i%16
    unsigned bsw = __builtin_amdgcn_permlanex16(b, b, 0x76543210u, 0xfedcba98u, false, false);
    unsigned asw = __builtin_amdgcn_permlanex16(a, a, 0x76543210u, 0xfedcba98u, false, false);
    a = (lane < 16) ? a : bsw;
    b = (lane < 16) ? asw : b;
#else
    unsigned blo = __shfl(b, (lane & 15), 32);
    unsigned ahi = __shfl(a, (lane & 15) + 16, 32);
    a = (lane < 16) ? a : blo;
    b = (lane < 16) ? ahi : b;
#endif
}

// Phase 1: 4 matrix ops, D(i,j) = A^T row-tile i x carry^T col-tile j (K=32)
__device__ __forceinline__ void mps_mul(const v16h a0, const v16h a1,
                                        const v16h B0, const v16h B1,
                                        U8 d[4]) {
    v8h cz;
#pragma unroll
    for (int k = 0; k < 8; ++k) cz[k] = (_Float16)0;
    d[0].h = __builtin_amdgcn_wmma_f16_16x16x32_f16(false, a0, false, B0, (short)0, cz, false, false);
    d[1].h = __builtin_amdgcn_wmma_f16_16x16x32_f16(false, a1, false, B0, (short)0, cz, false, false);
    d[2].h = __builtin_amdgcn_wmma_f16_16x16x32_f16(false, a0, false, B1, (short)0, cz, false, false);
    d[3].h = __builtin_amdgcn_wmma_f16_16x16x32_f16(false, a1, false, B1, (short)0, cz, false, false);
}

// Phase 2: re-form B operands from D tiles via half-wave swaps
__device__ __forceinline__ void mps_fix(U8 d[4], v16h &B0, v16h &B1, int lane) {
#pragma unroll
    for (int k = 0; k < 4; ++k) {
        swap_halves(d[0].u[k], d[1].u[k], lane);
        swap_halves(d[2].u[k], d[3].u[k], lane);
    }
    typedef unsigned uv8 __attribute__((ext_vector_type(8)));
    union { uv8 u; v16h h; } c0, c1;
#pragma unroll
    for (int k = 0; k < 4; ++k) {
        c0.u[k] = d[0].u[k]; c0.u[4 + k] = d[1].u[k];
        c1.u[k] = d[2].u[k]; c1.u[4 + k] = d[3].u[k];
    }
    B0 = c0.h;
    B1 = c1.h;
}

__global__ __launch_bounds__(WAVES_PER_BLOCK * 32)
void mps_chain_kernel(const float* __restrict__ s,
                      const float* __restrict__ A,
                      float* __restrict__ out) {
    __shared__ _Float16 lds[WAVES_PER_BLOCK][MPS_BOND * MPS_BOND];

    const int lane = threadIdx.x & 31;
    const int wave = threadIdx.x >> 5;
    const int b    = blockIdx.x * WAVES_PER_BLOCK + wave;
    const int half = lane >> 4;
    const int lm   = lane & 15;

    // ---- Loop-invariant A^T fragments in WMMA A-operand layout ----
    // elem e: K = (e<8 ? e : e+8) + 8*half, M = i*16 + lm; value = A[p][K][M]
    v16h afrag[2][2];
#pragma unroll
    for (int p = 0; p < 2; ++p)
#pragma unroll
        for (int i = 0; i < 2; ++i)
#pragma unroll
            for (int e = 0; e < 16; ++e) {
                int base = (e < 8) ? e : (e + 8);
                int K = base + 8 * half;
                int M = i * 16 + lm;
                afrag[p][i][e] = (_Float16)A[p * MPS_BOND * MPS_BOND + K * MPS_BOND + M];
            }

    // ---- two independent half-chain carries, both = I in B-operand layout ----
    // B elem e, in this code's contiguous k order (not the table order): K = e + 16*half, N_global = j*16 + lm
    v16h c1B0, c1B1, c2B0, c2B1;
#pragma unroll
    for (int e = 0; e < 16; ++e) {
        int K = e + 16 * half;
        _Float16 d0 = (_Float16)((K == lm) ? 1.0f : 0.0f);
        _Float16 d1 = (_Float16)((K == 16 + lm) ? 1.0f : 0.0f);
        c1B0[e] = d0; c1B1[e] = d1;
        c2B0[e] = d0; c2B1[e] = d1;
    }

    // ---- chain: 16 chunks x 32 sites; chain1 = sites [0,512), chain2 = [512,1024) ----
    const float* srow = s + (size_t)b * MPS_NSITES;
    for (int ch = 0; ch < MPS_HALFN / 32; ++ch) {
        float sv1 = srow[ch * 32 + lane];
        float sv2 = srow[MPS_HALFN + ch * 32 + lane];
        unsigned m1 = ballot32(sv1 > 0.0f);
        unsigned m2 = ballot32(sv2 > 0.0f);
#pragma unroll 4
        for (int t = 0; t < 32; ++t) {
            int b1 = __builtin_amdgcn_readfirstlane((int)((m1 >> t) & 1u));
            int b2 = __builtin_amdgcn_readfirstlane((int)((m2 >> t) & 1u));
            U8 d1[4], d2[4];
            // issue all 8 WMMAs before any dependent VALU (intra-wave ILP)
            if (b1) mps_mul(afrag[1][0], afrag[1][1], c1B0, c1B1, d1);
            else    mps_mul(afrag[0][0], afrag[0][1], c1B0, c1B1, d1);
            if (b2) mps_mul(afrag[1][0], afrag[1][1], c2B0, c2B1, d2);
            else    mps_mul(afrag[0][0], afrag[0][1], c2B0, c2B1, d2);
            mps_fix(d1, c1B0, c1B1, lane);
            mps_fix(d2, c2B0, c2B1, lane);
        }
    }

    // ---- trace(P) = trace(C2*C1) = sum_{a,b} C2[a][b] * C1[b][a] ----
    // store C1 (chain1 result) to per-wave LDS as plain row-major [K][N]
    _Float16* L = &lds[wave][0];
#pragma unroll
    for (int e = 0; e < 16; ++e) {
        int K = e + 16 * half;
        L[K * MPS_BOND + lm]      = c1B0[e];   // j=0: N = lm
        L[K * MPS_BOND + 16 + lm] = c1B1[e];   // j=1: N = 16+lm
    }
    // per-wave region, same-wave LDS ops are in order; compiler inserts waits
    float acc = 0.0f;
#pragma unroll
    for (int e = 0; e < 16; ++e) {
        int K = e + 16 * half;
        acc += (float)c2B0[e] * (float)L[lm * MPS_BOND + K];          // C2[K][lm]   * C1[lm][K]
        acc += (float)c2B1[e] * (float)L[(16 + lm) * MPS_BOND + K];   // C2[K][16+lm]* C1[16+lm][K]
    }
#pragma unroll
    for (int off = 16; off > 0; off >>= 1)
        acc += __shfl_xor(acc, off, 32);

    if (lane == 0) out[b] = logf(acc);
}

extern "C" void kernel_launch(void* const* d_in, const int* in_sizes, int n_in,
                              void* d_out, int out_size, void* d_ws, size_t ws_size,
                              hipStream_t stream) {
    const float* s = (const float*)d_in[0];   // [4096, 1024] f32
    const float* A = (const float*)d_in[1];   // [2, 32, 32]  f32
    float* out = (float*)d_out;               // [4096] f32
    (void)in_sizes; (void)n_in; (void)out_size; (void)d_ws; (void)ws_size;

    dim3 grid(MPS_BATCH / WAVES_PER_BLOCK);   // 512 blocks
    dim3 block(WAVES_PER_BLOCK * 32);         // 256 threads = 8 waves
    hipLaunchKernelGGL(mps_chain_kernel, grid, block, 0, stream, s, A, out);
}